// UnstructuredProbLoss_6923487281560
// MI455X (gfx1250) — compile-verified
//
#include <hip/hip_runtime.h>
#include <hip/hip_bf16.h>

// CDNA5 / gfx1250. logsumexp(dw_row + gw_row) == log(dot(exp dw_row, exp gw_row)),
// so the 677K-tuple partition functions become a 4096x4096x36 f32 GEMM on
// V_WMMA_F32_16X16X4_F32 (full f32 precision), then a per-tuple gather.
// Round 2: 4x4 register-blocked macro-tiles (4x fragment reuse) + wave-uniform
// culling of macro-tiles that no tuple can ever reference (~2x work cut).

typedef __attribute__((ext_vector_type(2))) float v2f;
typedef __attribute__((ext_vector_type(8))) float v8f;

#define NW 64
#define NN 4096          // NW*NW rows in each table
#define LD 32            // disc labels
#define KPAD 36          // 33 (incl. null) padded to multiple of 4
#define LC 64            // cont labels
#define MB 4             // M 16-tiles per wave (64 rows  = one i-block)
#define NB 4             // N 16-tiles per wave (64 cols = one k'-block)

// ---------------------------------------------------------------------------
// Prep: Ed = exp(dw - rowmax), Eg = exp(gw - rowmax), zero-padded to KPAD.
// ---------------------------------------------------------------------------
__global__ void prep_kernel(const float* __restrict__ dw,
                            const float* __restrict__ gw,
                            float* __restrict__ Ed, float* __restrict__ Eg,
                            float* __restrict__ maxd, float* __restrict__ maxg,
                            double* __restrict__ acc, float* __restrict__ out) {
  int idx = blockIdx.x * blockDim.x + threadIdx.x;
  if (idx == 0) { *acc = 0.0; *out = 0.0f; }
  if (idx >= 2 * NN) return;
  const float* src = (idx < NN) ? dw : gw;
  float* E        = (idx < NN) ? Ed : Eg;
  float* mx       = (idx < NN) ? maxd : maxg;
  int row = idx & (NN - 1);
  const float* w = src + (size_t)row * LD;
  float m = 0.0f;                       // null logit is 0
  #pragma unroll
  for (int l = 0; l < LD; ++l) m = fmaxf(m, w[l]);
  float* e = E + (size_t)row * KPAD;
  #pragma unroll
  for (int l = 0; l < LD; ++l) e[l] = expf(w[l] - m);
  e[LD]     = expf(-m);                 // null label
  e[LD + 1] = 0.0f; e[LD + 2] = 0.0f; e[LD + 3] = 0.0f;
  mx[row] = m;
}

// ---------------------------------------------------------------------------
// WMMA GEMM, 4x4 register-blocked: each wave computes a 64x64 macro-tile of
// logZ[row, col] = log( Ed[row,:] . Eg[col,:] ) + maxd[row] + maxg[col]
// for slab columns [c0, c0+W). K=36 via V_WMMA_F32_16X16X4_F32.
//
// Culling: rows of an M-macro-tile share one i (= row0/64); cols of an
// N-macro-tile share one k' (= colG0/64). Tuples require i+1 <= k' <= 62,
// so macro-tiles violating that are skipped (their logZ is never read).
// ---------------------------------------------------------------------------
__global__ void __launch_bounds__(256)
gemm_logz_kernel(const float* __restrict__ Ed, const float* __restrict__ Eg,
                 const float* __restrict__ maxd, const float* __restrict__ maxg,
                 float* __restrict__ logZ, int c0, int W) {
  const int lane  = threadIdx.x & 31;
  const int wave  = threadIdx.x >> 5;
  const int macro = blockIdx.x * 8 + wave;
  const int mm = macro & 63;            // 64 M-macro-tiles (4096 rows / 64)
  const int nm = macro >> 6;            // N-macro-tile within slab
  const int row0  = mm << 6;
  const int colL0 = nm << 6;
  const int colG0 = c0 + colL0;

  const int iblk   = mm;                // i value of all rows in this tile
  const int kprime = colG0 >> 6;        // k' value of all cols in this tile
  if (kprime <= iblk || kprime > NW - 2) return;   // no tuple maps here

  const int mlane = lane & 15;
  const int khalf = (lane >> 4) << 1;   // K sub-offset: 0 (lanes 0-15) / 2

  const float* ap[MB];
  const float* bp[NB];
  #pragma unroll
  for (int mb = 0; mb < MB; ++mb)
    ap[mb] = Ed + (size_t)(row0 + mb * 16 + mlane) * KPAD + khalf;
  #pragma unroll
  for (int nb = 0; nb < NB; ++nb)
    bp[nb] = Eg + (size_t)(colG0 + nb * 16 + mlane) * KPAD + khalf;

  v8f acc[MB][NB] = {};
  for (int k0 = 0; k0 < KPAD; k0 += 4) {   // 9 K-steps, kept rolled
    v2f a[MB], b[NB];
    #pragma unroll
    for (int mb = 0; mb < MB; ++mb) { a[mb].x = ap[mb][k0]; a[mb].y = ap[mb][k0 + 1]; }
    #pragma unroll
    for (int nb = 0; nb < NB; ++nb) { b[nb].x = bp[nb][k0]; b[nb].y = bp[nb][k0 + 1]; }
    #pragma unroll
    for (int mb = 0; mb < MB; ++mb)
      #pragma unroll
      for (int nb = 0; nb < NB; ++nb)
        acc[mb][nb] = __builtin_amdgcn_wmma_f32_16x16x4_f32(
            false, a[mb], false, b[nb], (short)0, acc[mb][nb], false, false);
  }

  // D layout: VGPR r -> lanes 0-15: M=r, N=lane; lanes 16-31: M=r+8, N=lane-16
  const int rowAdd = (lane >> 4) << 3;
  #pragma unroll
  for (int nb = 0; nb < NB; ++nb) {
    const int colL = colL0 + nb * 16 + mlane;
    const float mg = maxg[c0 + colL];
    #pragma unroll
    for (int mb = 0; mb < MB; ++mb) {
      #pragma unroll
      for (int r = 0; r < 8; ++r) {
        int mrow = row0 + mb * 16 + r + rowAdd;
        logZ[(size_t)mrow * W + colL] = logf(acc[mb][nb][r]) + maxd[mrow] + mg;
      }
    }
  }
}

// ---------------------------------------------------------------------------
// Continuous loss: per-cell logsumexp over 65 logits minus gold, upper tri.
// ---------------------------------------------------------------------------
__global__ void cont_kernel(const float* __restrict__ cw,
                            const int* __restrict__ gold,
                            double* __restrict__ acc) {
  __shared__ double sred[256];
  int idx = blockIdx.x * blockDim.x + threadIdx.x;   // 0..4095
  int i = idx >> 6, j = idx & 63;
  double v = 0.0;
  if (j >= i) {
    const float* w = cw + (size_t)idx * LC;
    float m = 0.0f;
    #pragma unroll 8
    for (int l = 0; l < LC; ++l) m = fmaxf(m, w[l]);
    float s = expf(-m);                              // null
    #pragma unroll 8
    for (int l = 0; l < LC; ++l) s += expf(w[l] - m);
    float lse = m + logf(s);
    int g = gold[idx];
    float gv = (g < LC) ? w[g] : 0.0f;
    v = (double)(lse - gv);
  }
  sred[threadIdx.x] = v;
  __syncthreads();
  for (int off = 128; off > 0; off >>= 1) {
    if (threadIdx.x < off) sred[threadIdx.x] += sred[threadIdx.x + off];
    __syncthreads();
  }
  if (threadIdx.x == 0) atomicAdd(acc, sred[0]);
}

// ---------------------------------------------------------------------------
// Gather: per tuple, loss = logZ[ci, di] - (dw[ci,g] + gw[di,g]).
// ---------------------------------------------------------------------------
__global__ void gather_kernel(const int* __restrict__ cont_idx,
                              const int* __restrict__ disc_idx,
                              const int* __restrict__ dgold,
                              const float* __restrict__ dw,
                              const float* __restrict__ gw,
                              const float* __restrict__ logZ,
                              int c0, int W, int T,
                              double* __restrict__ acc) {
  __shared__ double sred[256];
  int t = blockIdx.x * blockDim.x + threadIdx.x;
  double v = 0.0;
  if (t < T) {
    int di = disc_idx[t];
    if (di >= c0 && di < c0 + W) {
      int ci = cont_idx[t];
      int g  = dgold[t];
      float gv = (g < LD) ? (dw[(size_t)ci * LD + g] + gw[(size_t)di * LD + g])
                          : 0.0f;
      float lz = logZ[(size_t)ci * W + (di - c0)];
      v = (double)(lz - gv);
    }
  }
  sred[threadIdx.x] = v;
  __syncthreads();
  for (int off = 128; off > 0; off >>= 1) {
    if (threadIdx.x < off) sred[threadIdx.x] += sred[threadIdx.x + off];
    __syncthreads();
  }
  if (threadIdx.x == 0) atomicAdd(acc, sred[0]);
}

__global__ void finish_kernel(const double* __restrict__ acc,
                              float* __restrict__ out) {
  out[0] = (float)(*acc);
}

// ---------------------------------------------------------------------------
extern "C" void kernel_launch(void* const* d_in, const int* in_sizes, int n_in,
                              void* d_out, int out_size, void* d_ws,
                              size_t ws_size, hipStream_t stream) {
  (void)n_in; (void)out_size;
  const float* cont_w    = (const float*)d_in[0];   // [64,64,64]
  const float* disc_w    = (const float*)d_in[1];   // [64,64,32]
  const float* gap_w     = (const float*)d_in[2];   // [64,64,32]
  const int*   cont_gold = (const int*)d_in[3];     // [64,64]
  const int*   disc_gold = (const int*)d_in[4];     // [T]
  const int*   cont_idx  = (const int*)d_in[5];     // [T]
  const int*   disc_idx  = (const int*)d_in[6];     // [T]
  const int T = in_sizes[5];
  float* out = (float*)d_out;

  // Workspace layout (64B aligned regions).
  char* ws = (char*)d_ws;
  double* acc = (double*)ws;                                 // 8B
  float* maxd = (float*)(ws + 64);                           // 16KB
  float* maxg = (float*)(ws + 64 + 16384);                   // 16KB
  float* Ed   = (float*)(ws + 64 + 2 * 16384);               // 4096*36*4
  float* Eg   = Ed + (size_t)NN * KPAD;                      // 4096*36*4
  float* logZ = Eg + (size_t)NN * KPAD;                      // slab

  size_t fixed = 64 + 2 * 16384 + 2 * (size_t)NN * KPAD * sizeof(float);
  size_t avail = (ws_size > fixed) ? (ws_size - fixed) : 0;
  int W = (int)(avail / ((size_t)NN * sizeof(float)));       // slab columns
  W &= ~63;                       // macro-tiles are 64 columns wide
  if (W > NN) W = NN;
  if (W < 64) W = 64;  // minimum slab; assumes ws >= ~2.3MB

  prep_kernel<<<(2 * NN + 255) / 256, 256, 0, stream>>>(
      disc_w, gap_w, Ed, Eg, maxd, maxg, acc, out);
  cont_kernel<<<NN / 256, 256, 0, stream>>>(cont_w, cont_gold, acc);

  for (int c0 = 0; c0 < NN; c0 += W) {
    int Wc = (NN - c0 < W) ? (NN - c0) : W;    // multiple of 64
    int macros = 64 * (Wc / 64);               // M-macros * N-macros
    gemm_logz_kernel<<<macros / 8, 256, 0, stream>>>(
        Ed, Eg, maxd, maxg, logZ, c0, Wc);
    gather_kernel<<<(T + 255) / 256, 256, 0, stream>>>(
        cont_idx, disc_idx, disc_gold, disc_w, gap_w, logZ, c0, Wc, T, acc);
  }
  finish_kernel<<<1, 1, 0, stream>>>(acc, out);
}